// SparseMCFModel_12704513261941
// MI455X (gfx1250) — compile-verified
//
#include <hip/hip_runtime.h>
#include <hip/hip_bf16.h>
#include <math.h>

#define N_NODES 100000
#define N_EDGES 1600000
#define EL      (N_EDGES + N_NODES)   // edges + self loops
#define F_IN    128
#define DH      64
#define NH      4
#define NEG_SLOPE 0.2f
#define EPSF    1e-9f

typedef __attribute__((ext_vector_type(2))) float v2f;
typedef __attribute__((ext_vector_type(8))) float v8f;

// ---------------- WMMA f32 16x16x4 helper (CDNA5, wave32) ----------------
__device__ __forceinline__ v8f wmma4(v2f a, v2f b, v8f c) {
  // 8 args: (neg_a, A, neg_b, B, c_mod, C, reuse_a, reuse_b)
  return __builtin_amdgcn_wmma_f32_16x16x4_f32(false, a, false, b, (short)0, c,
                                               false, false);
}

// Ordered-key transform so unsigned atomicMax implements float max (incl. negatives)
__device__ __forceinline__ unsigned fkey(float f) {
  unsigned u = __float_as_uint(f);
  return (u & 0x80000000u) ? ~u : (u | 0x80000000u);
}
__device__ __forceinline__ float finv(unsigned k) {
  unsigned u = (k & 0x80000000u) ? (k ^ 0x80000000u) : ~k;
  return __uint_as_float(u);
}

__device__ __forceinline__ void atomAddF(float* p, float v) {
  unsafeAtomicAdd(p, v);   // native global_atomic_add_f32
}

__device__ __forceinline__ int selfSrc(const int* es, int i) {
  return (i < N_EDGES) ? es[i] : (i - N_EDGES);
}
__device__ __forceinline__ int selfDst(const int* ed, int i) {
  return (i < N_EDGES) ? ed[i] : (i - N_EDGES);
}

// ---------------- weight pack: interleave K-pairs for b64 B-loads --------
// P[(k/2)*C*2 + c*2 + (k&1)] = B[k*C + c]
__global__ void pack_b_kernel(const float* __restrict__ B, float* __restrict__ P,
                              int K, int C) {
  int idx = blockIdx.x * blockDim.x + threadIdx.x;
  if (idx >= K * C) return;
  int k = idx / C, c = idx % C;
  P[(k >> 1) * C * 2 + c * 2 + (k & 1)] = B[idx];
}

// W_gat (H,D,D) viewed as B[k][h*64+e] = W_gat[h][k][e], K=64, C=256
__global__ void pack_wgat_kernel(const float* __restrict__ Wgat,
                                 float* __restrict__ P) {
  int idx = blockIdx.x * blockDim.x + threadIdx.x;
  if (idx >= DH * NH * DH) return;
  int k = idx >> 8, col = idx & 255;
  int hh = col >> 6, e = col & 63;
  P[(k >> 1) * 512 + col * 2 + (k & 1)] = Wgat[hh * DH * DH + k * DH + e];
}

// ---------------- Encoder: h = relu([emb, dem] @ W_enc + b_enc) ----------
// grid.x = N/16 mtiles, block = (32,4): wave y handles ntile y (64 cols = 4 tiles)
// Main loop covers K=0..127 (emb); demand column K=128 folded in at store time.
__global__ void enc_kernel(const float* __restrict__ emb,
                           const float* __restrict__ dem,
                           const float* __restrict__ Wep,   // packed 128x64
                           const float* __restrict__ Wenc,  // raw (for row 128)
                           const float* __restrict__ benc,
                           float* __restrict__ h) {
  const int row0 = blockIdx.x * 16;
  const int col0 = threadIdx.y * 16;
  const int lane = threadIdx.x;
  const int half = lane >> 4;
  const int l15  = lane & 15;
  const int col  = col0 + l15;
  const float* arow = emb + (size_t)(row0 + l15) * F_IN;
  v8f acc = {};
  for (int k = 0; k < F_IN; k += 4) {
    const int ka = k + half * 2;
    const v2f a = *(const v2f*)(arow + ka);
    const v2f b = *(const v2f*)(Wep + (ka >> 1) * (DH * 2) + col * 2);
    acc = wmma4(a, b, acc);
  }
  const float wt = Wenc[F_IN * DH + col];   // demand-column weight
  const float bb = benc[col];
  for (int r = 0; r < 8; ++r) {
    const int row = row0 + half * 8 + r;
    h[row * DH + col] = fmaxf(acc[r] + dem[row] * wt + bb, 0.f);
  }
}

// ---------------- t = einsum('nd,hde->nhe'): N x 64 @ 64 x 256 -----------
// grid = (N/16, 4), block = (32,4): ntile = blockIdx.y*4 + threadIdx.y (16 total)
__global__ void t_kernel(const float* __restrict__ h,
                         const float* __restrict__ Wgp,   // packed 64x256
                         float* __restrict__ t) {
  const int row0  = blockIdx.x * 16;
  const int ntile = blockIdx.y * 4 + threadIdx.y;
  const int col0  = ntile * 16;
  const int lane  = threadIdx.x;
  const int half  = lane >> 4;
  const int l15   = lane & 15;
  const int col   = col0 + l15;
  const float* arow = h + (size_t)(row0 + l15) * DH;
  v8f acc = {};
  for (int k = 0; k < DH; k += 4) {
    const int ka = k + half * 2;
    const v2f a = *(const v2f*)(arow + ka);
    const v2f b = *(const v2f*)(Wgp + (ka >> 1) * 512 + col * 2);
    acc = wmma4(a, b, acc);
  }
  for (int r = 0; r < 8; ++r) {
    const int row = row0 + half * 8 + r;
    t[row * (NH * DH) + col] = acc[r];
  }
}

// ---------------- per-(node,head) attention logits ss, sd ----------------
__global__ void ssd_kernel(const float* __restrict__ t,
                           const float* __restrict__ asrc,
                           const float* __restrict__ adst,
                           float* __restrict__ ss, float* __restrict__ sd) {
  int idx = blockIdx.x * blockDim.x + threadIdx.x;
  if (idx >= N_NODES * NH) return;
  const int hh = idx & 3;
  const float* tp = t + (size_t)idx * DH;  // (n*4+hh)*64
  float s1 = 0.f, s2 = 0.f;
  #pragma unroll 8
  for (int e = 0; e < DH; ++e) {
    float v = tp[e];
    s1 += v * asrc[hh * DH + e];
    s2 += v * adst[hh * DH + e];
  }
  ss[idx] = s1;
  sd[idx] = s2;
}

// ---------------- edge scores + segment max (by src) ---------------------
__global__ void escore_kernel(const int* __restrict__ es, const int* __restrict__ ed,
                              const float* __restrict__ ss, const float* __restrict__ sd,
                              float* __restrict__ ev, unsigned* __restrict__ mkeys) {
  int idx = blockIdx.x * blockDim.x + threadIdx.x;
  if (idx >= EL * NH) return;
  const int i = idx >> 2, hh = idx & 3;
  const int s = selfSrc(es, i), d = selfDst(ed, i);
  float v = ss[s * NH + hh] + sd[d * NH + hh];
  v = (v >= 0.f) ? v : NEG_SLOPE * v;           // leaky relu
  ev[idx] = v;
  atomicMax(&mkeys[s * NH + hh], fkey(v));
}

// ---------------- exp(e - max) + segment sum -----------------------------
__global__ void expsum_kernel(const int* __restrict__ es,
                              float* __restrict__ ev,
                              const unsigned* __restrict__ mkeys,
                              float* __restrict__ den) {
  int idx = blockIdx.x * blockDim.x + threadIdx.x;
  if (idx >= EL * NH) return;
  const int i = idx >> 2, hh = idx & 3;
  const int s = selfSrc(es, i);
  const float m = finv(mkeys[s * NH + hh]);
  const float x = expf(ev[idx] - m);
  ev[idx] = x;
  atomAddF(&den[s * NH + hh], x);
}

// ---------------- aggregation: agg[src] += alpha * t[dst] ----------------
// one thread per (edge, float4-chunk); 64 chunks of 4 cover the 256 (h,e)
__global__ void agg_kernel(const int* __restrict__ es, const int* __restrict__ ed,
                           const float* __restrict__ ev, const float* __restrict__ den,
                           const float* __restrict__ t, float* __restrict__ agg) {
  int tid = blockIdx.x * blockDim.x + threadIdx.x;
  if (tid >= EL * 64) return;
  const int i  = tid >> 6;
  const int j  = tid & 63;
  const int hh = j >> 4;
  const int s = selfSrc(es, i), d = selfDst(ed, i);
  const float alpha = ev[i * NH + hh] / (den[s * NH + hh] + EPSF);
  const float4 tv = *(const float4*)(t + (size_t)d * (NH * DH) + j * 4);
  float* ap = agg + (size_t)s * (NH * DH) + j * 4;
  atomAddF(ap + 0, alpha * tv.x);
  atomAddF(ap + 1, alpha * tv.y);
  atomAddF(ap + 2, alpha * tv.z);
  atomAddF(ap + 3, alpha * tv.w);
}

// ---------------- nxt = relu(mean_h(agg) + b_gat) ------------------------
__global__ void nxt_kernel(const float* __restrict__ agg,
                           const float* __restrict__ bgat,
                           float* __restrict__ nxt) {
  int tid = blockIdx.x * blockDim.x + threadIdx.x;
  if (tid >= N_NODES * DH) return;
  const int n = tid >> 6, e = tid & 63;
  const float* ap = agg + (size_t)n * (NH * DH);
  float v = (ap[e] + ap[DH + e] + ap[2 * DH + e] + ap[3 * DH + e]) * 0.25f + bgat[e];
  nxt[tid] = fmaxf(v, 0.f);
}

// ---------------- GRU gate: hout = z*nxt + (1-z)*h, z = sig(nxt@Wgx + h@Wgh + bg)
// grid.x = N/16, block = (32,4): 4 ntiles of 16 cover 64 cols
__global__ void gru_kernel(const float* __restrict__ nxt,
                           const float* __restrict__ hin,
                           const float* __restrict__ Wgxp,  // packed 64x64
                           const float* __restrict__ Wghp,  // packed 64x64
                           const float* __restrict__ bg,
                           float* __restrict__ hout) {
  const int row0 = blockIdx.x * 16;
  const int col0 = threadIdx.y * 16;
  const int lane = threadIdx.x;
  const int half = lane >> 4;
  const int l15  = lane & 15;
  const int col  = col0 + l15;
  const float* nrow = nxt + (size_t)(row0 + l15) * DH;
  const float* hrow = hin + (size_t)(row0 + l15) * DH;
  v8f acc = {};
  for (int k = 0; k < DH; k += 4) {       // nxt @ W_gx
    const int ka = k + half * 2;
    const v2f a = *(const v2f*)(nrow + ka);
    const v2f b = *(const v2f*)(Wgxp + (ka >> 1) * (DH * 2) + col * 2);
    acc = wmma4(a, b, acc);
  }
  for (int k = 0; k < DH; k += 4) {       // + h @ W_gh (same accumulator)
    const int ka = k + half * 2;
    const v2f a = *(const v2f*)(hrow + ka);
    const v2f b = *(const v2f*)(Wghp + (ka >> 1) * (DH * 2) + col * 2);
    acc = wmma4(a, b, acc);
  }
  const float bb = bg[col];
  for (int r = 0; r < 8; ++r) {
    const int row = row0 + half * 8 + r;
    const float z  = 1.f / (1.f + expf(-(acc[r] + bb)));
    const float nv = nxt[row * DH + col];
    const float hv = hin[row * DH + col];
    hout[row * DH + col] = z * nv + (1.f - z) * hv;
  }
}

// ---------------- decoder: pred = h @ W_dec + b_dec ----------------------
__global__ void dec_kernel(const float* __restrict__ h,
                           const float* __restrict__ Wdec,
                           const float* __restrict__ bdec,
                           float* __restrict__ pred) {
  int n = blockIdx.x * blockDim.x + threadIdx.x;
  if (n >= N_NODES) return;
  const float4* hp = (const float4*)(h + (size_t)n * DH);
  const float4* wp = (const float4*)Wdec;
  float s = 0.f;
  #pragma unroll
  for (int q = 0; q < DH / 4; ++q) {
    float4 a = hp[q], w = wp[q];
    s += a.x * w.x + a.y * w.y + a.z * w.z + a.w * w.w;
  }
  pred[n] = s + bdec[0];
}

// ---------------- fw segment softmax over edge_src (E edges) -------------
__global__ void fwmax_kernel(const int* __restrict__ es, const float* __restrict__ pred,
                             unsigned* __restrict__ mk) {
  int i = blockIdx.x * blockDim.x + threadIdx.x;
  if (i >= N_EDGES) return;
  const int s = es[i];
  atomicMax(&mk[s], fkey(pred[s]));
}
__global__ void fwsum_kernel(const int* __restrict__ es, const float* __restrict__ pred,
                             const unsigned* __restrict__ mk, float* __restrict__ den) {
  int i = blockIdx.x * blockDim.x + threadIdx.x;
  if (i >= N_EDGES) return;
  const int s = es[i];
  atomAddF(&den[s], expf(pred[s] - finv(mk[s])));
}
__global__ void fw_kernel(const int* __restrict__ es, const float* __restrict__ pred,
                          const unsigned* __restrict__ mk, const float* __restrict__ den,
                          float* __restrict__ fw) {
  int i = blockIdx.x * blockDim.x + threadIdx.x;
  if (i >= N_EDGES) return;
  const int s = es[i];
  fw[i] = expf(pred[s] - finv(mk[s])) / (den[s] + EPSF);
}

// ---------------- flow iteration ----------------------------------------
__global__ void scatter_kernel(const int* __restrict__ ed, const float* __restrict__ fv,
                               float* __restrict__ inflow) {
  int i = blockIdx.x * blockDim.x + threadIdx.x;
  if (i >= N_EDGES) return;
  atomAddF(&inflow[ed[i]], fv[i]);
}
__global__ void flowmap_kernel(const int* __restrict__ es, const float* __restrict__ fw,
                               const float* __restrict__ inflow, const float* __restrict__ dem,
                               float* __restrict__ fout) {
  int i = blockIdx.x * blockDim.x + threadIdx.x;
  if (i >= N_EDGES) return;
  const int s = es[i];
  fout[i] = fw[i] * fmaxf(inflow[s] - dem[s], 0.f);
}

// ---------------- cost = sum(flow^2) ------------------------------------
__global__ void cost_kernel(const float* __restrict__ flow, float* __restrict__ out) {
  __shared__ float red[256];
  float s = 0.f;
  for (int i = blockIdx.x * blockDim.x + threadIdx.x; i < N_EDGES;
       i += gridDim.x * blockDim.x) {
    float f = flow[i];
    s += f * f;
  }
  red[threadIdx.x] = s;
  __syncthreads();
  for (int off = 128; off > 0; off >>= 1) {
    if (threadIdx.x < off) red[threadIdx.x] += red[threadIdx.x + off];
    __syncthreads();
  }
  if (threadIdx.x == 0) atomAddF(out, red[0]);
}

// =========================================================================
extern "C" void kernel_launch(void* const* d_in, const int* in_sizes, int n_in,
                              void* d_out, int out_size, void* d_ws, size_t ws_size,
                              hipStream_t stream) {
  const float* dem  = (const float*)d_in[0];
  const float* emb  = (const float*)d_in[1];
  const int*   esrc = (const int*)d_in[2];
  const int*   edst = (const int*)d_in[3];
  const float* Wenc = (const float*)d_in[4];
  const float* benc = (const float*)d_in[5];
  const float* Wgat = (const float*)d_in[6];
  const float* asrc = (const float*)d_in[7];
  const float* adst = (const float*)d_in[8];
  const float* bgat = (const float*)d_in[9];
  const float* Wgx  = (const float*)d_in[10];
  const float* Wgh  = (const float*)d_in[11];
  const float* bg   = (const float*)d_in[12];
  const float* Wdec = (const float*)d_in[13];
  const float* bdec = (const float*)d_in[14];

  float* out  = (float*)d_out;
  float* flow = out + 1;
  float* fw   = out + 1 + N_EDGES;

  // workspace carve (all sizes 256B-aligned)
  char* p = (char*)d_ws;
  auto carve = [&p](size_t bytes) -> void* {
    void* r = (void*)p;
    p += (bytes + 255) & ~(size_t)255;
    return r;
  };
  float*    h0     = (float*)carve((size_t)N_NODES * DH * 4);
  float*    h1     = (float*)carve((size_t)N_NODES * DH * 4);
  float*    t      = (float*)carve((size_t)N_NODES * NH * DH * 4);
  float*    ss     = (float*)carve((size_t)N_NODES * NH * 4);
  float*    sd     = (float*)carve((size_t)N_NODES * NH * 4);
  float*    ev     = (float*)carve((size_t)EL * NH * 4);
  unsigned* mkeys  = (unsigned*)carve((size_t)N_NODES * NH * 4);
  float*    den    = (float*)carve((size_t)N_NODES * NH * 4);
  float*    agg    = (float*)carve((size_t)N_NODES * NH * DH * 4);
  float*    nxtbuf = (float*)carve((size_t)N_NODES * DH * 4);
  float*    pred   = (float*)carve((size_t)N_NODES * 4);
  unsigned* mk2    = (unsigned*)carve((size_t)N_NODES * 4);
  float*    den2   = (float*)carve((size_t)N_NODES * 4);
  float*    inflow = (float*)carve((size_t)N_NODES * 4);
  float*    fvA    = (float*)carve((size_t)N_EDGES * 4);
  float*    fvB    = (float*)carve((size_t)N_EDGES * 4);
  float*    Wep    = (float*)carve((size_t)F_IN * DH * 4);        // packed W_enc[0:128]
  float*    Wgp    = (float*)carve((size_t)DH * NH * DH * 4);     // packed W_gat
  float*    Wgxp   = (float*)carve((size_t)DH * DH * 4);
  float*    Wghp   = (float*)carve((size_t)DH * DH * 4);

  const int MT = N_NODES / 16;                 // 6250 exact
  const dim3 wblk(32, 4);
  const int B256 = 256;
  const int gNH   = (N_NODES * NH + B256 - 1) / B256;
  const int gELH  = (EL * NH + B256 - 1) / B256;
  const int gAGG  = (EL * 64 + B256 - 1) / B256;
  const int gND   = (N_NODES * DH + B256 - 1) / B256;
  const int gN    = (N_NODES + B256 - 1) / B256;
  const int gE    = (N_EDGES + B256 - 1) / B256;

  // ---- pack weights into WMMA-friendly K-pair interleave ----
  pack_b_kernel<<<(F_IN * DH + 255) / 256, B256, 0, stream>>>(Wenc, Wep, F_IN, DH);
  pack_wgat_kernel<<<(DH * NH * DH + 255) / 256, B256, 0, stream>>>(Wgat, Wgp);
  pack_b_kernel<<<(DH * DH + 255) / 256, B256, 0, stream>>>(Wgx, Wgxp, DH, DH);
  pack_b_kernel<<<(DH * DH + 255) / 256, B256, 0, stream>>>(Wgh, Wghp, DH, DH);

  // ---- encoder ----
  enc_kernel<<<MT, wblk, 0, stream>>>(emb, dem, Wep, Wenc, benc, h0);

  // ---- 2 GAT layers (ping-pong h0 <-> h1; final result back in h0) ----
  float* hin = h0;
  float* hout = h1;
  for (int layer = 0; layer < 2; ++layer) {
    t_kernel<<<dim3(MT, 4), wblk, 0, stream>>>(hin, Wgp, t);
    ssd_kernel<<<gNH, B256, 0, stream>>>(t, asrc, adst, ss, sd);
    hipMemsetAsync(mkeys, 0, (size_t)N_NODES * NH * 4, stream);
    escore_kernel<<<gELH, B256, 0, stream>>>(esrc, edst, ss, sd, ev, mkeys);
    hipMemsetAsync(den, 0, (size_t)N_NODES * NH * 4, stream);
    expsum_kernel<<<gELH, B256, 0, stream>>>(esrc, ev, mkeys, den);
    hipMemsetAsync(agg, 0, (size_t)N_NODES * NH * DH * 4, stream);
    agg_kernel<<<gAGG, B256, 0, stream>>>(esrc, edst, ev, den, t, agg);
    nxt_kernel<<<gND, B256, 0, stream>>>(agg, bgat, nxtbuf);
    gru_kernel<<<MT, wblk, 0, stream>>>(nxtbuf, hin, Wgxp, Wghp, bg, hout);
    float* tmp = hin; hin = hout; hout = tmp;
  }
  // after 2 swaps, final h is in `hin` == h0

  // ---- decoder + fw softmax over edge_src ----
  dec_kernel<<<gN, B256, 0, stream>>>(hin, Wdec, bdec, pred);
  hipMemsetAsync(mk2, 0, (size_t)N_NODES * 4, stream);
  fwmax_kernel<<<gE, B256, 0, stream>>>(esrc, pred, mk2);
  hipMemsetAsync(den2, 0, (size_t)N_NODES * 4, stream);
  fwsum_kernel<<<gE, B256, 0, stream>>>(esrc, pred, mk2, den2);
  fw_kernel<<<gE, B256, 0, stream>>>(esrc, pred, mk2, den2, fw);

  // ---- 10 flow iterations ----
  const float* cur = fw;
  float* pong[2] = {fvA, fvB};
  for (int it = 0; it < 10; ++it) {
    hipMemsetAsync(inflow, 0, (size_t)N_NODES * 4, stream);
    scatter_kernel<<<gE, B256, 0, stream>>>(edst, cur, inflow);
    float* nxtp = (it == 9) ? flow : pong[it & 1];
    flowmap_kernel<<<gE, B256, 0, stream>>>(esrc, fw, inflow, dem, nxtp);
    cur = nxtp;
  }

  // ---- cost = sum(flow^2) ----
  hipMemsetAsync(out, 0, 4, stream);
  cost_kernel<<<1024, B256, 0, stream>>>(flow, out);
}